// PointwiseConvBlurModel_87136296501462
// MI455X (gfx1250) — compile-verified
//
#include <hip/hip_runtime.h>
#include <stdint.h>

// Problem constants (match reference)
#define KSZ   17          // blur kernel size
#define HP    256         // patch H = W
#define IW    272         // padded img H = W (256 + 16)
#define P2    65536       // HP*HP pixels
#define KK    289         // KSZ*KSZ taps
#define BCN   12          // B*C planes sharing each per-pixel kernel

// Tiling
#define TW    64          // tile width (pixels)
#define TH    8           // tile height (pixels)
#define LCOLS 80          // TW + 16 halo cols
#define LROWS 16          // TH + max(khg) - 1 = 8 + 9 - 1
#define PLANE (LROWS * LCOLS)   // 1280 floats per img plane in LDS
#define NTHREADS 256

// ---------------------------------------------------------------------------
// Zero d_out (harness poisons it; each output element then gets exactly two
// commutative f32 atomic adds -> bitwise deterministic result).
// ---------------------------------------------------------------------------
__global__ __launch_bounds__(256) void pcb_zero_out(float4* __restrict__ out4) {
    out4[blockIdx.x * 256 + threadIdx.x] = float4{0.f, 0.f, 0.f, 0.f};
}

// ---------------------------------------------------------------------------
// Main spatially-varying blur.
// grid = 256 blocks: [2 kh-groups] x [4 x-tiles] x [32 y-tiles]
// block = 256 threads = 8 wave32; each wave owns one tile row (32 lanes x 2px).
// ---------------------------------------------------------------------------
__global__ __launch_bounds__(NTHREADS) void pcb_blur_kernel(
    const float* __restrict__ img,      // (12, 272, 272)
    const float* __restrict__ kernels,  // (16, 1, 289, 65536)
    const int*   __restrict__ idxp,     // scalar
    float*       __restrict__ out)      // (12, 256, 256), pre-zeroed
{
    __shared__ float sm[BCN * PLANE];   // 61,440 B

    int b = blockIdx.x;
    const int g   = b & 1;              // kh-group: 0 -> kh 0..8, 1 -> kh 9..16
    b >>= 1;
    const int tx0 = (b & 3) * TW;       // 0,64,128,192
    const int ty0 = (b >> 2) * TH;      // 0..248

    const int kh0   = g ? 9 : 0;
    const int kh1   = g ? 17 : 9;
    const int khg   = kh1 - kh0;        // 9 or 8
    const int nrows = TH + khg - 1;     // 16 or 15 img rows needed

    const int tid = threadIdx.x;
    const int gy0 = ty0 + kh0;          // first img row this group touches

    // ---- Stage img halo tile into LDS with CDNA5 async global->LDS b128 ----
    // 12 planes x nrows x 80 cols; both global and LDS addresses 16B aligned.
    const int total = BCN * nrows * (LCOLS / 4);
    for (int e = tid; e < total; e += NTHREADS) {
        const int c4 = e % (LCOLS / 4);
        const int r  = (e / (LCOLS / 4)) % nrows;
        const int pl = e / ((LCOLS / 4) * nrows);
        const float* gp = img + (size_t)(pl * IW + gy0 + r) * IW + tx0 + c4 * 4;
        // generic shared pointer: low 32 bits are the LDS byte offset
        unsigned lds_addr = (unsigned)(unsigned long long)(const void*)
                            &sm[pl * PLANE + r * LCOLS + c4 * 4];
        unsigned long long ga = (unsigned long long)gp;
        asm volatile("global_load_async_to_lds_b128 %0, %1, off"
                     :: "v"(lds_addr), "v"(ga) : "memory");
    }
    asm volatile("s_wait_asynccnt 0" ::: "memory");
    __syncthreads();

    // ---- Main multiply-accumulate over taps ----
    const int lane = tid & 31;          // 32 lanes span one tile row
    const int wy   = tid >> 5;          // 0..7 tile row
    const int px   = lane * 2;          // local x of this thread's pixel pair

    const int idx = *idxp;
    const float* kb = kernels + (size_t)idx * ((size_t)KK * P2)
                    + (size_t)(ty0 + wy) * HP + tx0 + px;

    float acc0[BCN], acc1[BCN];
#pragma unroll
    for (int bc = 0; bc < BCN; ++bc) { acc0[bc] = 0.f; acc1[bc] = 0.f; }

    for (int kh = kh0; kh < kh1; ++kh) {
        // per-pixel kernel taps for this kh row: 17 coalesced b64 loads/wave
        float2 kr[KSZ];
#pragma unroll
        for (int kw = 0; kw < KSZ; ++kw)
            kr[kw] = *(const float2*)(kb + (size_t)(kh * KSZ + kw) * P2);

        const int lrow = wy + (kh - kh0);
        const float* vp = &sm[lrow * LCOLS + px];
#pragma unroll
        for (int bc = 0; bc < BCN; ++bc) {
            // 18-float sliding window from LDS (9 aligned ds_load_b64)
            float v[KSZ + 1];
#pragma unroll
            for (int j = 0; j < 9; ++j) {
                float2 t = *(const float2*)(vp + bc * PLANE + 2 * j);
                v[2 * j]     = t.x;
                v[2 * j + 1] = t.y;
            }
#pragma unroll
            for (int kw = 0; kw < KSZ; ++kw) {
                acc0[bc] = fmaf(kr[kw].x, v[kw],     acc0[bc]);
                acc1[bc] = fmaf(kr[kw].y, v[kw + 1], acc1[bc]);
            }
        }
    }

    // ---- Accumulate partials: exactly 2 f32 atomic adds per element ----
    float* ob = out + (size_t)(ty0 + wy) * HP + tx0 + px;
#pragma unroll
    for (int bc = 0; bc < BCN; ++bc) {
        atomicAdd(ob + (size_t)bc * P2,     acc0[bc]);
        atomicAdd(ob + (size_t)bc * P2 + 1, acc1[bc]);
    }
}

// ---------------------------------------------------------------------------
extern "C" void kernel_launch(void* const* d_in, const int* in_sizes, int n_in,
                              void* d_out, int out_size, void* d_ws, size_t ws_size,
                              hipStream_t stream) {
    const float* img  = (const float*)d_in[0];
    const float* kern = (const float*)d_in[1];
    const int*   idxp = (const int*)d_in[2];
    float* out = (float*)d_out;

    // 12*256*256 = 786,432 floats = 196,608 float4 = 768 blocks exactly
    pcb_zero_out<<<768, 256, 0, stream>>>((float4*)out);
    pcb_blur_kernel<<<256, NTHREADS, 0, stream>>>(img, kern, idxp, out);
}